// VariableLengthMemoryLayer_43645457662548
// MI455X (gfx1250) — compile-verified
//
#include <hip/hip_runtime.h>
#include <cfloat>

#define HOPS 3
#define BDIM 4096
#define L    200
#define D    64
#define STR  68          // padded LDS row stride in floats: 272B (16B aligned, bank step 4)
#define NT   256
#define LTILES 13        // ceil(200/16)

typedef float v2f __attribute__((ext_vector_type(2)));
typedef float v8f __attribute__((ext_vector_type(8)));
typedef int   v4i __attribute__((ext_vector_type(4)));

#if __has_builtin(__builtin_amdgcn_global_load_async_to_lds_b128)
#define HAVE_ASYNC_LDS 1
// Probe-derived signature: (AS1 v4i* global_src, AS3 v4i* lds_dst, imm offset, imm cpol)
typedef __attribute__((address_space(1))) v4i GVec;
typedef __attribute__((address_space(3))) v4i LVec;
#else
#define HAVE_ASYNC_LDS 0
#endif

static __device__ __forceinline__ void wait_async_lds() {
#if __has_builtin(__builtin_amdgcn_s_wait_asynccnt)
    __builtin_amdgcn_s_wait_asynccnt(0);
#else
    asm volatile("s_wait_asynccnt 0" ::: "memory");
#endif
}

// ---- dynamic LDS layout (floats) ----
#define OFF_MEM   0
#define OFF_OM    (L * STR)                 // 13600
#define OFF_SC    (2 * L * STR)             // 27200
#define OFF_ATTN  (OFF_SC + 208)
#define OFF_Z     (OFF_ATTN + 208)
#define OFF_PREV  (OFF_Z + D)
#define OFF_PART  (OFF_PREV + D)
#define OFF_RED   (OFF_PART + 256)
#define OFF_BC    (OFF_RED + 8)
#define SMEM_FLOATS (OFF_BC + 2)
#define SMEM_BYTES  (SMEM_FLOATS * sizeof(float))

__global__ void vlml_kernel(const float* __restrict__ uq,
                            const float* __restrict__ iq,
                            const float* __restrict__ gmem,
                            const float* __restrict__ gom,
                            const float* __restrict__ W,
                            const float* __restrict__ bias,
                            const long long* __restrict__ seq,
                            float* __restrict__ out) {
    extern __shared__ float smem[];
    float* s_mem  = smem + OFF_MEM;   // [200][STR]
    float* s_om   = smem + OFF_OM;    // [200][STR]
    float* s_sc   = smem + OFF_SC;    // [208] raw scores
    float* s_attn = smem + OFF_ATTN;  // [200] softmax weights
    float* s_z    = smem + OFF_Z;     // [64]
    float* s_prev = smem + OFF_PREV;  // [64]
    float* s_part = smem + OFF_PART;  // [4][64] partial out sums
    float* s_red  = smem + OFF_RED;   // [8] per-wave reduction
    float* s_bc   = smem + OFF_BC;    // [2] broadcast (max, sum)

    const int b    = blockIdx.x;
    const int tid  = threadIdx.x;
    const int lane = tid & 31;
    const int wave = tid >> 5;

    const float* gm = gmem + (size_t)b * (L * D);
    const float* go = gom  + (size_t)b * (L * D);

    // ---- phase 0: stage memory[b] and output_memory[b] into LDS (async DMA path) ----
    for (int i = tid; i < (L * D) / 4; i += NT) {
        const int row = i >> 4;           // 16 float4 chunks per 64-float row
        const int c   = (i & 15) << 2;
#if HAVE_ASYNC_LDS
        __builtin_amdgcn_global_load_async_to_lds_b128(
            (GVec*)const_cast<float*>(gm + row * D + c),
            (LVec*)(s_mem + row * STR + c), 0, 0);
        __builtin_amdgcn_global_load_async_to_lds_b128(
            (GVec*)const_cast<float*>(go + row * D + c),
            (LVec*)(s_om + row * STR + c), 0, 0);
#else
        *(float4*)(s_mem + row * STR + c) = *(const float4*)(gm + row * D + c);
        *(float4*)(s_om  + row * STR + c) = *(const float4*)(go + row * D + c);
#endif
    }
    if (tid < D) s_z[tid] = uq[b * D + tid] + iq[b * D + tid];
    const long long sl = seq[b];
#if HAVE_ASYNC_LDS
    wait_async_lds();
#endif
    __syncthreads();

    for (int h = 0; h < HOPS; ++h) {
        // ---- z update (hops 1,2): z = relu(z @ W[h-1]^T + bias[h-1] + prev_out) ----
        if (h > 0) {
            float acc = 0.f;
            if (tid < D) {
                const float* wrow = W + ((size_t)(h - 1) * D + tid) * D;
                acc = bias[(h - 1) * D + tid] + s_prev[tid];
#pragma unroll 16
                for (int k = 0; k < D; ++k) acc += s_z[k] * wrow[k];
            }
            __syncthreads();
            if (tid < D) s_z[tid] = fmaxf(acc, 0.f);
            __syncthreads();
        }

        // ---- scores = memory[b] @ z via V_WMMA_F32_16X16X4_F32 ----
        // B operand: z replicated across all 16 N columns; K mapping mirrors the
        // documented A layout (VGPR0 -> K 0|2 per lane-half, VGPR1 -> K 1|3).
        {
            const int m  = lane & 15;
            const int kh = (lane >> 4) << 1;     // 0 for lanes 0-15, 2 for lanes 16-31
            for (int t = wave; t < LTILES; t += 8) {   // wave-uniform trip: EXEC stays full
                const int l0 = t << 4;
                v8f c = {};
                const float* arow = s_mem + (l0 + m) * STR;
#pragma unroll
                for (int k0 = 0; k0 < D; k0 += 4) {
                    v2f a, bb;
                    a.x  = arow[k0 + kh];
                    a.y  = arow[k0 + kh + 1];
                    bb.x = s_z[k0 + kh];
                    bb.y = s_z[k0 + kh + 1];
                    c = __builtin_amdgcn_wmma_f32_16x16x4_f32(
                            false, a, false, bb, (short)0, c, false, false);
                }
                // C layout: VGPR r holds M=r (lanes 0-15) and M=8+r (lanes 16-31);
                // all N columns identical -> lane 0 / lane 16 publish 8 rows each.
                if (lane == 0) {
#pragma unroll
                    for (int r = 0; r < 8; ++r) s_sc[l0 + r] = c[r];
                } else if (lane == 16) {
#pragma unroll
                    for (int r = 0; r < 8; ++r) s_sc[l0 + 8 + r] = c[r];
                }
            }
        }
        __syncthreads();

        // ---- masked softmax over l < 200 (mask: l < seq_length -> keep, else fmin) ----
        float x = -FLT_MAX;
        if (tid < L) x = ((long long)tid < sl) ? s_sc[tid] : -FLT_MAX;
        float m = x;
#pragma unroll
        for (int o = 16; o > 0; o >>= 1) m = fmaxf(m, __shfl_xor(m, o, 32));
        if (lane == 0) s_red[wave] = m;
        __syncthreads();
        if (tid == 0) {
            float mm = s_red[0];
#pragma unroll
            for (int w = 1; w < 8; ++w) mm = fmaxf(mm, s_red[w]);
            s_bc[0] = mm;
        }
        __syncthreads();
        const float mx = s_bc[0];
        float e = (tid < L) ? expf(x - mx) : 0.f;
        float s = e;
#pragma unroll
        for (int o = 16; o > 0; o >>= 1) s += __shfl_xor(s, o, 32);
        if (lane == 0) s_red[wave] = s;
        __syncthreads();
        if (tid == 0) {
            float ss = 0.f;
#pragma unroll
            for (int w = 0; w < 8; ++w) ss += s_red[w];
            s_bc[1] = ss;
        }
        __syncthreads();
        const float inv = 1.f / s_bc[1];
        if (tid < L) {
            const float a = e * inv;
            s_attn[tid] = a;
            out[(size_t)h * BDIM * L + (size_t)b * L + tid] = a;   // attns[h,b,l]
        }
        __syncthreads();

        // ---- out = attn @ output_memory[b]  (4 l-segments of 50, then combine) ----
        {
            const int d = tid & 63;
            const int g = tid >> 6;           // 0..3
            float acc = 0.f;
            const int l0 = g * 50;
#pragma unroll 10
            for (int l = l0; l < l0 + 50; ++l) acc += s_attn[l] * s_om[l * STR + d];
            s_part[g * 64 + d] = acc;
        }
        __syncthreads();
        if (tid < D) {
            const float o4 = s_part[tid] + s_part[64 + tid] + s_part[128 + tid] + s_part[192 + tid];
            s_prev[tid] = o4;
            out[(size_t)HOPS * BDIM * L + (size_t)h * BDIM * D + (size_t)b * D + tid] = o4; // outs[h,b,d]
        }
        __syncthreads();
    }
}

extern "C" void kernel_launch(void* const* d_in, const int* in_sizes, int n_in,
                              void* d_out, int out_size, void* d_ws, size_t ws_size,
                              hipStream_t stream) {
    const float*     uq   = (const float*)d_in[0];
    const float*     iq   = (const float*)d_in[1];
    const float*     mem  = (const float*)d_in[2];
    const float*     om   = (const float*)d_in[3];
    const float*     W    = (const float*)d_in[4];
    const float*     bias = (const float*)d_in[5];
    const long long* seq  = (const long long*)d_in[6];
    float* outp = (float*)d_out;

    // ~112 KB dynamic LDS per block: opt in above the default cap.
    (void)hipFuncSetAttribute((const void*)vlml_kernel,
                              hipFuncAttributeMaxDynamicSharedMemorySize,
                              (int)SMEM_BYTES);

    vlml_kernel<<<dim3(BDIM), dim3(NT), SMEM_BYTES, stream>>>(
        uq, iq, mem, om, W, bias, seq, outp);
}